// InternalGraphConvolutionLayer_38019050504902
// MI455X (gfx1250) — compile-verified
//
#include <hip/hip_runtime.h>
#include <hip/hip_bf16.h>

// CDNA5 / gfx1250, wave32. Fused GCN layer:
//   out = relu(x@W + segment_sum((x@M)[edge_src], edge_dst))
// Kernel 1: dual fp32 WMMA GEMM (self term -> d_out, msg -> d_ws)
// Kernel 2: coalesced edge scatter with global_atomic_add_f32
// Kernel 3: in-place float4 ReLU

typedef __attribute__((ext_vector_type(2))) float v2f;
typedef __attribute__((ext_vector_type(8))) float v8f;

#define DIM 256
#define LDS_STRIDE 260   // pad: 260*4B = 1040B row stride, conflict-free on 64x4B banks

__global__ __launch_bounds__(256) void gcn_dual_gemm_wmma(
    const float* __restrict__ x,
    const float* __restrict__ W,
    const float* __restrict__ M,
    float* __restrict__ selfOut,   // d_out: x@W (pre-relu)
    float* __restrict__ msgOut,    // d_ws : x@M
    int nNodes)
{
    __shared__ float lds_a[16 * LDS_STRIDE];  // 16.6 KB of the 320 KB WGP LDS

    const int row0 = blockIdx.x * 16;
    const int tid  = threadIdx.x;          // 256 threads = 8 waves (wave32)
    const int wave = tid >> 5;
    const int lane = tid & 31;

    // Stage 16x256 fp32 strip of x into LDS (each thread: 16 floats as 4 x float4).
    {
        const int r    = tid >> 4;         // 0..15 (row in strip)
        const int c0   = (tid & 15) * 16;  // 0,16,...,240
        const int grow = row0 + r;
        const bool ok  = (grow < nNodes);
        const float* src = x + (size_t)grow * DIM + c0;
        float* dst = lds_a + r * LDS_STRIDE + c0;   // 1040B stride keeps 16B align
#pragma unroll
        for (int i = 0; i < 4; ++i) {
            float4 v = ok ? *(const float4*)(src + 4 * i)
                          : make_float4(0.f, 0.f, 0.f, 0.f);
            *(float4*)(dst + 4 * i) = v;
        }
    }
    __syncthreads();

    const int laneHi = lane >> 4;      // 0 or 1 (selects K pair within fragment)
    const int laneLo = lane & 15;      // M (for A/C) or N (for B/C)

    // Each wave owns two adjacent 16-col output tiles for BOTH weight matrices.
    const int nA = wave * 32 + laneLo; // column for tile 0
    const int nB = nA + 16;            // column for tile 1

    v8f accW0 = {}, accW1 = {}, accM0 = {}, accM1 = {};

    for (int k0 = 0; k0 < DIM; k0 += 4) {
        const int kb = k0 + laneHi * 2;

        // A fragment (shared by all 4 WMMAs): row laneLo, K = kb, kb+1
        const float* ap = lds_a + laneLo * LDS_STRIDE + kb;
        v2f a; a.x = ap[0]; a.y = ap[1];

        // B fragments: B[kb..kb+1][n] row-major, stride DIM (L2/WGP$-resident)
        v2f bW0, bW1, bM0, bM1;
        bW0.x = W[(size_t)kb * DIM + nA];       bW0.y = W[(size_t)(kb + 1) * DIM + nA];
        bW1.x = W[(size_t)kb * DIM + nB];       bW1.y = W[(size_t)(kb + 1) * DIM + nB];
        bM0.x = M[(size_t)kb * DIM + nA];       bM0.y = M[(size_t)(kb + 1) * DIM + nA];
        bM1.x = M[(size_t)kb * DIM + nB];       bM1.y = M[(size_t)(kb + 1) * DIM + nB];

        accW0 = __builtin_amdgcn_wmma_f32_16x16x4_f32(false, a, false, bW0, (short)0, accW0, false, false);
        accW1 = __builtin_amdgcn_wmma_f32_16x16x4_f32(false, a, false, bW1, (short)0, accW1, false, false);
        accM0 = __builtin_amdgcn_wmma_f32_16x16x4_f32(false, a, false, bM0, (short)0, accM0, false, false);
        accM1 = __builtin_amdgcn_wmma_f32_16x16x4_f32(false, a, false, bM1, (short)0, accM1, false, false);
    }

    // D layout: VGPR r -> (M = r + 8*laneHi, N = laneLo)
#pragma unroll
    for (int r = 0; r < 8; ++r) {
        const int m = row0 + r + 8 * laneHi;
        if (m < nNodes) {
            selfOut[(size_t)m * DIM + nA] = accW0[r];
            selfOut[(size_t)m * DIM + nB] = accW1[r];
            msgOut [(size_t)m * DIM + nA] = accM0[r];
            msgOut [(size_t)m * DIM + nB] = accM1[r];
        }
    }
}

// 64 threads per edge, 4 floats per thread: a wave32 covers 512 contiguous bytes
// of one edge's message vector -> fully coalesced gather + atomic scatter.
__global__ __launch_bounds__(256) void gcn_edge_scatter(
    const float* __restrict__ msg,
    const int* __restrict__ esrc,
    const int* __restrict__ edst,
    float* __restrict__ out,
    int nEdges)
{
    const long long gid = (long long)blockIdx.x * 256 + threadIdx.x;
    const int e = (int)(gid >> 6);
    if (e >= nEdges) return;
    const int c = (int)(gid & 63) * 4;

    const int s = esrc[e];
    const int d = edst[e];

    const float4 v = *(const float4*)(msg + (size_t)s * DIM + c);
    float* p = out + (size_t)d * DIM + c;
    atomicAdd(p + 0, v.x);   // global_atomic_add_f32, non-returning (STOREcnt)
    atomicAdd(p + 1, v.y);
    atomicAdd(p + 2, v.z);
    atomicAdd(p + 3, v.w);
}

__global__ __launch_bounds__(256) void gcn_relu4(float* __restrict__ out, int nVec4)
{
    const int i = blockIdx.x * 256 + threadIdx.x;
    if (i >= nVec4) return;
    float4 v = ((float4*)out)[i];
    v.x = fmaxf(v.x, 0.f);
    v.y = fmaxf(v.y, 0.f);
    v.z = fmaxf(v.z, 0.f);
    v.w = fmaxf(v.w, 0.f);
    ((float4*)out)[i] = v;
}

extern "C" void kernel_launch(void* const* d_in, const int* in_sizes, int n_in,
                              void* d_out, int out_size, void* d_ws, size_t ws_size,
                              hipStream_t stream) {
    const float* x    = (const float*)d_in[0];
    const float* W    = (const float*)d_in[1];
    const float* M    = (const float*)d_in[2];
    const int*   esrc = (const int*)d_in[3];
    const int*   edst = (const int*)d_in[4];
    float* out = (float*)d_out;
    float* msg = (float*)d_ws;                 // needs nNodes*256*4 = 10.24 MB

    const int nNodes = in_sizes[0] / DIM;      // 10000
    const int nEdges = in_sizes[3];            // 320000

    // 1) self = x@W -> d_out (full overwrite = init), msg = x@M -> d_ws
    const int rowTiles = (nNodes + 15) / 16;
    gcn_dual_gemm_wmma<<<rowTiles, 256, 0, stream>>>(x, W, M, out, msg, nNodes);

    // 2) out[dst] += msg[src] for each edge
    const long long scatterThreads = (long long)nEdges * 64;
    const int scatterBlocks = (int)((scatterThreads + 255) / 256);
    gcn_edge_scatter<<<scatterBlocks, 256, 0, stream>>>(msg, esrc, edst, out, nEdges);

    // 3) out = relu(out)
    const int nVec4 = out_size / 4;
    gcn_relu4<<<(nVec4 + 255) / 256, 256, 0, stream>>>(out, nVec4);
}